// ClusterLoss_77403900608667
// MI455X (gfx1250) — compile-verified
//
#include <hip/hip_runtime.h>
#include <math.h>

// ---------------------------------------------------------------------------
// Problem constants (match reference)
// ---------------------------------------------------------------------------
#define N_PTS   262144
#define DIM     128
#define KCL     256
#define ALPHA_C 0.1f

// ---------------------------------------------------------------------------
// Workspace layout (floats)
// ---------------------------------------------------------------------------
#define OFF_WSUM   0                        // [K*D]  weighted sums
#define OFF_ESUM   (KCL*DIM)                // [K*D]  unweighted sums
#define OFF_MSUM   (2*KCL*DIM)              // [K]    sum of m
#define OFF_CNT    (2*KCL*DIM + KCL)        // [K]    counts (float)
#define OFF_SQ     (2*KCL*DIM + 2*KCL)      // [K]    sum of ||e||^2
#define OFF_ACC    (2*KCL*DIM + 3*KCL)      // [2]    {intra_acc, inter_acc}
#define OFF_CEN    (OFF_ACC + 4)            // [K*D]  centroids
#define OFF_CNORM  (OFF_CEN + KCL*DIM)      // [K]    ||c||^2
#define OFF_Q      (OFF_CNORM + KCL)        // [K]    sizes^0.25
#define ZERO_FLOATS (OFF_ACC + 2)           // region that must start at zero
#define LDS_FLOATS  (2*KCL*DIM + 3*KCL)     // 66304 floats = 259 KB (fits 320KB WGP LDS)

typedef __attribute__((ext_vector_type(2))) float v2f;
typedef __attribute__((ext_vector_type(8))) float v8f;

__device__ __forceinline__ float waveReduceAdd(float v) {
    // wave32 butterfly reduction
    v += __shfl_xor(v, 16, 32);
    v += __shfl_xor(v, 8, 32);
    v += __shfl_xor(v, 4, 32);
    v += __shfl_xor(v, 2, 32);
    v += __shfl_xor(v, 1, 32);
    return v;
}

// Force native HW fp32 atomic add on global memory (global_atomic_add_f32).
__device__ __forceinline__ void gAtomicAdd(float* p, float v) {
    unsafeAtomicAdd(p, v);
}

// ---------------------------------------------------------------------------
// Kernel 0: zero the accumulator region of the workspace
// ---------------------------------------------------------------------------
__global__ void cl_zero_kernel(float* __restrict__ g) {
    int idx = blockIdx.x * blockDim.x + threadIdx.x;
    if (idx < ZERO_FLOATS) g[idx] = 0.0f;
}

// ---------------------------------------------------------------------------
// Kernel 1: streaming pass over all points.
//   LDS-privatized segment sums (wsum, esum, msum, cnt, sqsum), then one
//   atomic flush per block to global accumulators.
//   One wave handles one point at a time: 512B coalesced row load
//   (float4 per lane), ds_add_f32 scatter into LDS (<=2-way bank conflict).
// ---------------------------------------------------------------------------
__global__ void __launch_bounds__(1024)
cl_pass1_kernel(const float* __restrict__ emb,
                const int*   __restrict__ labels,
                const float* __restrict__ mass,
                float* __restrict__ g) {
    extern __shared__ float smem[];   // LDS_FLOATS floats, same layout as g[0..)

    const int tid = threadIdx.x;
    // zero LDS accumulators
    for (int i = tid; i < LDS_FLOATS; i += blockDim.x) smem[i] = 0.0f;
    __syncthreads();

    const int lane        = tid & 31;
    const int waveInBlk   = tid >> 5;
    const int wavesPerBlk = blockDim.x >> 5;
    const int gwave       = blockIdx.x * wavesPerBlk + waveInBlk;
    const int totalWaves  = gridDim.x * wavesPerBlk;

    for (int n = gwave; n < N_PTS; n += totalWaves) {
        const int   lab = labels[n];              // broadcast load
        const float mm  = sqrtf(mass[n]);         // mass ** 0.5
        const float4 e  = ((const float4*)(emb + (size_t)n * DIM))[lane];

        const int base = lab * DIM + lane * 4;    // bank = (4*lane+i) & 63 -> 2-way
        atomicAdd(&smem[OFF_WSUM + base + 0], e.x * mm);
        atomicAdd(&smem[OFF_WSUM + base + 1], e.y * mm);
        atomicAdd(&smem[OFF_WSUM + base + 2], e.z * mm);
        atomicAdd(&smem[OFF_WSUM + base + 3], e.w * mm);
        atomicAdd(&smem[OFF_ESUM + base + 0], e.x);
        atomicAdd(&smem[OFF_ESUM + base + 1], e.y);
        atomicAdd(&smem[OFF_ESUM + base + 2], e.z);
        atomicAdd(&smem[OFF_ESUM + base + 3], e.w);

        float s = e.x * e.x + e.y * e.y + e.z * e.z + e.w * e.w;
        s = waveReduceAdd(s);                     // ||e_n||^2
        if (lane == 0) {
            atomicAdd(&smem[OFF_MSUM + lab], mm);
            atomicAdd(&smem[OFF_CNT  + lab], 1.0f);
            atomicAdd(&smem[OFF_SQ   + lab], s);
        }
    }
    __syncthreads();

    // flush block-private accumulators to global (identical layout)
    for (int i = tid; i < LDS_FLOATS; i += blockDim.x) {
        float v = smem[i];
        if (v != 0.0f) gAtomicAdd(&g[i], v);
    }
}

// ---------------------------------------------------------------------------
// Kernel 2: per-cluster centroid + intra-loss term + cnorm + q.
//   One wave per cluster (256 waves total).
//   intra_k = sqsum_k - 2*c.esum + cnt*||c||^2 ;  acc += intra_k/(cnt*K)
// ---------------------------------------------------------------------------
__global__ void __launch_bounds__(256)
cl_centroid_kernel(const int* __restrict__ sizes, float* __restrict__ g) {
    const int lane = threadIdx.x & 31;
    const int k    = (blockIdx.x * blockDim.x + threadIdx.x) >> 5;  // cluster id

    const float  ms  = g[OFF_MSUM + k];
    const float4 ws  = ((const float4*)(g + OFF_WSUM))[k * (DIM / 4) + lane];
    const float4 es  = ((const float4*)(g + OFF_ESUM))[k * (DIM / 4) + lane];
    const float  inv = 1.0f / ms;
    float4 c;
    c.x = ws.x * inv; c.y = ws.y * inv; c.z = ws.z * inv; c.w = ws.w * inv;
    ((float4*)(g + OFF_CEN))[k * (DIM / 4) + lane] = c;

    float ce = c.x * es.x + c.y * es.y + c.z * es.z + c.w * es.w;
    float cc = c.x * c.x + c.y * c.y + c.z * c.z + c.w * c.w;
    ce = waveReduceAdd(ce);
    cc = waveReduceAdd(cc);

    if (lane == 0) {
        const float cnt   = g[OFF_CNT + k];
        const float intra = g[OFF_SQ + k] - 2.0f * ce + cnt * cc;
        gAtomicAdd(&g[OFF_ACC + 0], intra / (cnt * (float)KCL));
        g[OFF_CNORM + k] = cc;
        g[OFF_Q + k]     = sqrtf(sqrtf((float)sizes[k]));  // sizes ** 0.25
    }
}

// ---------------------------------------------------------------------------
// Kernel 3: inter-cluster loss via f32 WMMA Gram matrix G = C * C^T.
//   136 upper-triangular 16x16 tiles, one wave each, 32 x V_WMMA_F32_16X16X4_F32.
//   ISA layouts (wave32):
//     A 16x4 : lane<16 -> M=lane, VGPR0=K0,VGPR1=K1 ; lane>=16 -> K2,K3
//     B 4x16 : lane<16 -> N=lane, VGPR0=K0,VGPR1=K1 ; lane>=16 -> K2,K3
//     C/D    : VGPR j, lane l -> M = j + 8*(l>>4), N = l&15
// ---------------------------------------------------------------------------
__global__ void __launch_bounds__(32)
cl_gram_kernel(float* __restrict__ g) {
    const int lane = threadIdx.x;      // 0..31, all lanes active (EXEC all 1s)
    const int lm   = lane & 15;
    const int half = lane >> 4;

    // decode linear tile id -> (ti, tj) with ti <= tj  (uniform scalar loop)
    int rem = blockIdx.x, ti = 0;
    while (rem >= (16 - ti)) { rem -= (16 - ti); ++ti; }
    const int tj      = ti + rem;
    const int rowBase = ti * 16;
    const int colBase = tj * 16;

    const float* __restrict__ cen = g + OFF_CEN;
    const float* Arow = cen + (size_t)(rowBase + lm) * DIM;
    const float* Brow = cen + (size_t)(colBase + lm) * DIM;
    const int koff = 2 * half;

    v8f acc = {};
    #pragma unroll
    for (int d0 = 0; d0 < DIM; d0 += 4) {
        v2f a, b;
        a.x = Arow[d0 + koff]; a.y = Arow[d0 + koff + 1];
        b.x = Brow[d0 + koff]; b.y = Brow[d0 + koff + 1];
        acc = __builtin_amdgcn_wmma_f32_16x16x4_f32(
                false, a, false, b, (short)0, acc, false, false);
    }

    // d^2(i,j) = ||c_i||^2 + ||c_j||^2 - 2 G_ij ; sum q_i q_j / sqrt(d^2)
    const int   gj = colBase + lm;
    const float qj = g[OFF_Q + gj];
    const float nj = g[OFF_CNORM + gj];
    float sum = 0.0f;
    #pragma unroll
    for (int j = 0; j < 8; ++j) {
        const int gi = rowBase + j + 8 * half;
        if (gi < gj) {
            const float d2 = fmaxf(g[OFF_CNORM + gi] + nj - 2.0f * acc[j], 1e-24f);
            sum += g[OFF_Q + gi] * qj * __frsqrt_rn(d2);
        }
    }
    sum = waveReduceAdd(sum);
    if (lane == 0) {
        const float invPairs = 2.0f / ((float)KCL * (float)(KCL - 1));  // 1/32640
        gAtomicAdd(&g[OFF_ACC + 1], sum * invPairs);
    }
}

// ---------------------------------------------------------------------------
// Kernel 4: finalize -> out = {loss, loss_intra, ALPHA*loss_inter}
// ---------------------------------------------------------------------------
__global__ void cl_final_kernel(const float* __restrict__ g, float* __restrict__ out) {
    if (threadIdx.x == 0 && blockIdx.x == 0) {
        const float li = g[OFF_ACC + 0];
        const float le = ALPHA_C * g[OFF_ACC + 1];
        out[0] = li + le;
        out[1] = li;
        out[2] = le;
    }
}

// ---------------------------------------------------------------------------
extern "C" void kernel_launch(void* const* d_in, const int* in_sizes, int n_in,
                              void* d_out, int out_size, void* d_ws, size_t ws_size,
                              hipStream_t stream) {
    const float* emb    = (const float*)d_in[0];
    const int*   labels = (const int*)d_in[1];
    const float* mass   = (const float*)d_in[2];
    const int*   sizes  = (const int*)d_in[3];
    float*       out    = (float*)d_out;
    float*       g      = (float*)d_ws;

    // 0) zero accumulators
    cl_zero_kernel<<<(ZERO_FLOATS + 255) / 256, 256, 0, stream>>>(g);

    // 1) streaming segment sums, LDS-privatized (259 KB dynamic LDS per WG)
    const int    blocks1 = 256;
    const size_t ldsB    = (size_t)LDS_FLOATS * sizeof(float);
    cl_pass1_kernel<<<blocks1, 1024, ldsB, stream>>>(emb, labels, mass, g);

    // 2) centroids + intra loss (one wave per cluster)
    cl_centroid_kernel<<<KCL / 8, 256, 0, stream>>>(sizes, g);

    // 3) WMMA Gram + inter loss (136 upper-triangular 16x16 tiles)
    cl_gram_kernel<<<136, 32, 0, stream>>>(g);

    // 4) finalize
    cl_final_kernel<<<1, 32, 0, stream>>>(g, out);
}